// GaussianAvatar_14594298872337
// MI455X (gfx1250) — compile-verified
//
#include <hip/hip_runtime.h>
#include <math.h>

// Problem constants (match reference)
constexpr int B_ = 32, V_ = 25000, J_ = 20, F_ = 50000, N_ = 200000;
constexpr float MIN_SCALE = 0.0005f, MAX_SCALE = 0.05f;

typedef float v2f __attribute__((ext_vector_type(2)));
typedef float v8f __attribute__((ext_vector_type(8)));

// ---------------------------------------------------------------------------
// Kernel 0: per-batch joint matrices. One thread per batch (32 threads).
// Computes Rodrigues rotations, chains the kinematic transforms, and writes
// Mpad[b, j, 0..15]: 3x4 matrix [s*Rg | s*tg + trans] row-major in cols 0..11,
// cols 12..15 zero (padding so one WMMA N-tile == one batch).
// Folding trans is exact because skin weights sum to 1.
// ---------------------------------------------------------------------------
__global__ void k0_joint_mats(const float* __restrict__ pose,
                              const float* __restrict__ trans,
                              const float* __restrict__ scale,
                              const float* __restrict__ jrest,
                              float* __restrict__ Mpad) {
  int b = threadIdx.x;
  if (b >= B_) return;
  const float* p = pose + b * (J_ * 3);
  float s = scale[b];
  float trx = trans[b * 3 + 0], try_ = trans[b * 3 + 1], trz = trans[b * 3 + 2];

  // G = identity (rot), zero translation
  float G[3][4] = {{1.f, 0.f, 0.f, 0.f}, {0.f, 1.f, 0.f, 0.f}, {0.f, 0.f, 1.f, 0.f}};

  for (int j = 0; j < J_; ++j) {
    // Rodrigues: R = c*I + s*K + (1-c)*a a^T
    float ax = p[3 * j + 0], ay = p[3 * j + 1], az = p[3 * j + 2];
    float th = sqrtf(ax * ax + ay * ay + az * az + 1e-12f);
    float inv = 1.0f / th;
    float x = ax * inv, y = ay * inv, z = az * inv;
    float sn = sinf(th), cs = cosf(th), oc = 1.0f - cs;
    float R[3][3] = {
        {cs + oc * x * x, oc * x * y - sn * z, oc * x * z + sn * y},
        {oc * x * y + sn * z, cs + oc * y * y, oc * y * z - sn * x},
        {oc * x * z - sn * y, oc * y * z + sn * x, cs + oc * z * z}};

    // local translation: jrest[0] for j==0 else jrest[j]-jrest[j-1]
    float lt[3];
    for (int c = 0; c < 3; ++c)
      lt[c] = (j == 0) ? jrest[c] : (jrest[3 * j + c] - jrest[3 * (j - 1) + c]);

    // G = G o [R | lt]
    float NG[3][4];
    for (int i = 0; i < 3; ++i) {
      for (int c = 0; c < 3; ++c)
        NG[i][c] = G[i][0] * R[0][c] + G[i][1] * R[1][c] + G[i][2] * R[2][c];
      NG[i][3] = G[i][0] * lt[0] + G[i][1] * lt[1] + G[i][2] * lt[2] + G[i][3];
    }
    for (int i = 0; i < 3; ++i)
      for (int c = 0; c < 4; ++c) G[i][c] = NG[i][c];

    // A_j = [Rg | Gt - Rg*jrest[j]], folded with scale & trans
    float jr0 = jrest[3 * j + 0], jr1 = jrest[3 * j + 1], jr2 = jrest[3 * j + 2];
    float* out = Mpad + (size_t)(b * J_ + j) * 16;
    for (int i = 0; i < 3; ++i) {
      float tg = G[i][3] - (G[i][0] * jr0 + G[i][1] * jr1 + G[i][2] * jr2);
      out[i * 4 + 0] = s * G[i][0];
      out[i * 4 + 1] = s * G[i][1];
      out[i * 4 + 2] = s * G[i][2];
      out[i * 4 + 3] = s * tg + (i == 0 ? trx : (i == 1 ? try_ : trz));
    }
    out[12] = 0.f; out[13] = 0.f; out[14] = 0.f; out[15] = 0.f;
  }
}

// ---------------------------------------------------------------------------
// Zero-fill scratch (normal accumulator) -- d_ws is poisoned by the harness.
// ---------------------------------------------------------------------------
__global__ void k_zero(float* __restrict__ p, int n) {
  int i = blockIdx.x * blockDim.x + threadIdx.x;
  if (i < n) p[i] = 0.f;
}

// ---------------------------------------------------------------------------
// Kernel 1: WMMA skinning.  One wave per (16-vertex tile, batch).
// C(16x16) = W(16x20) @ Mpad_b(20x16) via five V_WMMA_F32_16X16X4_F32 steps.
// Operand layouts per ISA 7.12.2 (f32):
//   A 16x4 : VGPR0 = {lanes0-15: K=0, lanes16-31: K=2}, VGPR1 = {K=1, K=3}
//   B 4x16 : VGPR0 = {lanes0-15: row K=0, lanes16-31: row K=2}, VGPR1 = {K=1,K=3}
//   C 16x16: VGPR r = {lanes0-15: M=r, lanes16-31: M=r+8}, N = lane%16
// All ten operand fragments are fetched branchlessly up front (one load
// clause, one wait), then the five WMMAs issue back-to-back.
// Epilogue stages C through LDS and applies [v_template;1] -> Vw[b,v,0:3].
// ---------------------------------------------------------------------------
__global__ void __launch_bounds__(32)
k1_skin_wmma(const float* __restrict__ w,     // (V, J)
             const float* __restrict__ Mpad,  // (B, J, 16)
             const float* __restrict__ vt,    // (V, 3)
             float* __restrict__ Vw) {        // (B, V, 3)
  __shared__ float tile[16][17];  // [M (vertex)][N (matrix entry)], padded
  const int lane = threadIdx.x;
  const int row = lane & 15;
  const int half = lane >> 4;
  const int v0 = blockIdx.x * 16;
  const int b = blockIdx.y;

  const int vrow = v0 + row;
  const bool valid = (vrow < V_);
  const int vclamp = valid ? vrow : (V_ - 1);  // branchless: clamp + zero
  const float amask = valid ? 1.f : 0.f;       // zeroed rows -> C rows 0

  const float* arow = w + (size_t)vclamp * J_ + 2 * half;   // 8B-aligned
  const float* brow = Mpad + (size_t)(b * J_ + 2 * half) * 16 + row;

  v2f afrag[5], bfrag[5];
#pragma unroll
  for (int k = 0; k < 5; ++k) {
    v2f a = *(const v2f*)(arow + 4 * k);  // {j0, j0+1}, j0 = 4k + 2*half
    afrag[k].x = a.x * amask;
    afrag[k].y = a.y * amask;
    bfrag[k].x = brow[4 * k * 16];        // row j0,   col 'row'
    bfrag[k].y = brow[(4 * k + 1) * 16];  // row j0+1, col 'row'
  }

  v8f acc = {0.f, 0.f, 0.f, 0.f, 0.f, 0.f, 0.f, 0.f};
#pragma unroll
  for (int k = 0; k < 5; ++k) {
    acc = __builtin_amdgcn_wmma_f32_16x16x4_f32(
        /*neg_a=*/false, afrag[k], /*neg_b=*/false, bfrag[k],
        /*c_mod=*/(short)0, acc, /*reuse_a=*/false, /*reuse_b=*/false);
  }

#pragma unroll
  for (int r = 0; r < 8; ++r) tile[r + 8 * half][row] = acc[r];
  __syncthreads();

  if (half == 0 && valid) {
    const float* T = tile[row];  // blended 3x4 (row-major, entries 0..11)
    float px = vt[3 * vrow + 0], py = vt[3 * vrow + 1], pz = vt[3 * vrow + 2];
    float ox = T[0] * px + T[1] * py + T[2] * pz + T[3];
    float oy = T[4] * px + T[5] * py + T[6] * pz + T[7];
    float oz = T[8] * px + T[9] * py + T[10] * pz + T[11];
    float* o = Vw + ((size_t)b * V_ + vrow) * 3;
    o[0] = ox; o[1] = oy; o[2] = oz;
  }
}

// ---------------------------------------------------------------------------
// Kernel 2: face normals scatter-added into per-vertex accumulator.
// ---------------------------------------------------------------------------
__global__ void k2_face_normals(const float* __restrict__ Vw,
                                const int* __restrict__ faces,
                                float* __restrict__ nrm) {
  int f = blockIdx.x * blockDim.x + threadIdx.x;
  int b = blockIdx.y;
  if (f >= F_) return;
  int i0 = faces[3 * f + 0], i1 = faces[3 * f + 1], i2 = faces[3 * f + 2];
  const float* Vb = Vw + (size_t)b * V_ * 3;
  float ax = Vb[3 * i0], ay = Vb[3 * i0 + 1], az = Vb[3 * i0 + 2];
  float e1x = Vb[3 * i1] - ax, e1y = Vb[3 * i1 + 1] - ay, e1z = Vb[3 * i1 + 2] - az;
  float e2x = Vb[3 * i2] - ax, e2y = Vb[3 * i2 + 1] - ay, e2z = Vb[3 * i2 + 2] - az;
  float fx = e1y * e2z - e1z * e2y;
  float fy = e1z * e2x - e1x * e2z;
  float fz = e1x * e2y - e1y * e2x;
  float* Nb = nrm + (size_t)b * V_ * 3;
  atomicAdd(&Nb[3 * i0 + 0], fx); atomicAdd(&Nb[3 * i0 + 1], fy); atomicAdd(&Nb[3 * i0 + 2], fz);
  atomicAdd(&Nb[3 * i1 + 0], fx); atomicAdd(&Nb[3 * i1 + 1], fy); atomicAdd(&Nb[3 * i1 + 2], fz);
  atomicAdd(&Nb[3 * i2 + 0], fx); atomicAdd(&Nb[3 * i2 + 1], fy); atomicAdd(&Nb[3 * i2 + 2], fz);
}

// ---------------------------------------------------------------------------
// Kernel 3: per-Gaussian streaming pass (the bandwidth-bound one: 358 MB out).
// Gathers Vw / accumulated normal (both L2-resident), builds the tangent
// frame inline, rot->quat, quat-mul, activations, writes all 5 outputs.
// ---------------------------------------------------------------------------
__global__ void __launch_bounds__(256)
k3_gauss(const float* __restrict__ Vw, const float* __restrict__ nrm,
         const float* __restrict__ off, const float* __restrict__ colors_raw,
         const float* __restrict__ opacity_raw, const float* __restrict__ log_scales,
         const float* __restrict__ quats, const int* __restrict__ vidx,
         float* __restrict__ o_means, float* __restrict__ o_colors,
         float* __restrict__ o_opacity, float* __restrict__ o_scales,
         float* __restrict__ o_rot) {
  int n = blockIdx.x * blockDim.x + threadIdx.x;
  int b = blockIdx.y;
  if (n >= N_) return;

  int vi = vidx[n];
  size_t vb = ((size_t)b * V_ + vi) * 3;
  float px = Vw[vb], py = Vw[vb + 1], pz = Vw[vb + 2];

  // n = normalize(accumulated normal)
  float ax = nrm[vb], ay = nrm[vb + 1], az = nrm[vb + 2];
  float ninv = 1.f / sqrtf(ax * ax + ay * ay + az * az + 1e-12f);
  float nx = ax * ninv, ny = ay * ninv, nz = az * ninv;

  // t = normalize(ref - (ref.n)n), ref = |n.x|<0.9 ? x-axis : z-axis
  float rx, ry, rz;
  if (fabsf(nx) < 0.9f) { rx = 1.f; ry = 0.f; rz = 0.f; }
  else                  { rx = 0.f; ry = 0.f; rz = 1.f; }
  float d = rx * nx + ry * ny + rz * nz;
  float tx = rx - d * nx, ty = ry - d * ny, tz = rz - d * nz;
  float tinv = 1.f / sqrtf(tx * tx + ty * ty + tz * tz + 1e-12f);
  tx *= tinv; ty *= tinv; tz *= tinv;

  // bt = cross(n, t)
  float bx = ny * tz - nz * ty;
  float by = nz * tx - nx * tz;
  float bz = nx * ty - ny * tx;

  // means = base + ox*t + oy*bt + oz*n
  float ox = off[3 * n], oy = off[3 * n + 1], oz = off[3 * n + 2];
  float mx = px + ox * tx + oy * bx + oz * nx;
  float my = py + ox * ty + oy * by + oz * ny;
  float mz = pz + ox * tz + oy * bz + oz * nz;

  // frame = [t | bt | n] (columns) -> quaternion (reference branch order)
  float r00 = tx, r01 = bx, r02 = nx;
  float r10 = ty, r11 = by, r12 = ny;
  float r20 = tz, r21 = bz, r22 = nz;
  float tr = r00 + r11 + r22;
  float qw, qx, qy, qz;
  if (tr > 0.f) {
    float s1 = sqrtf(fmaxf(tr + 1.f, 1e-8f)) * 2.f;
    qw = 0.25f * s1; qx = (r21 - r12) / s1; qy = (r02 - r20) / s1; qz = (r10 - r01) / s1;
  } else if ((r00 > r11) && (r00 > r22)) {
    float s2 = sqrtf(fmaxf(1.f + r00 - r11 - r22, 1e-8f)) * 2.f;
    qw = (r21 - r12) / s2; qx = 0.25f * s2; qy = (r01 + r10) / s2; qz = (r02 + r20) / s2;
  } else if (r11 > r22) {
    float s3 = sqrtf(fmaxf(1.f + r11 - r00 - r22, 1e-8f)) * 2.f;
    qw = (r02 - r20) / s3; qx = (r01 + r10) / s3; qy = 0.25f * s3; qz = (r12 + r21) / s3;
  } else {
    float s4 = sqrtf(fmaxf(1.f + r22 - r00 - r11, 1e-8f)) * 2.f;
    qw = (r10 - r01) / s4; qx = (r02 + r20) / s4; qy = (r12 + r21) / s4; qz = 0.25f * s4;
  }
  float qn = 1.f / sqrtf(qw * qw + qx * qx + qy * qy + qz * qz + 1e-12f);
  qw *= qn; qx *= qn; qy *= qn; qz *= qn;

  // lq = normalize(quaternions[n]); rot = quat_mul(fq, lq)
  float4 lqv = *(const float4*)(quats + 4 * (size_t)n);  // 16B-aligned
  float lw = lqv.x, lx = lqv.y, ly = lqv.z, lz = lqv.w;
  float lninv = 1.f / sqrtf(lw * lw + lx * lx + ly * ly + lz * lz + 1e-12f);
  lw *= lninv; lx *= lninv; ly *= lninv; lz *= lninv;
  float rw = qw * lw - qx * lx - qy * ly - qz * lz;
  float rxq = qw * lx + qx * lw + qy * lz - qz * ly;
  float ryq = qw * ly - qx * lz + qy * lw + qz * lx;
  float rzq = qw * lz + qx * ly - qy * lx + qz * lw;

  size_t gi = (size_t)b * N_ + n;
  o_means[3 * gi + 0] = mx; o_means[3 * gi + 1] = my; o_means[3 * gi + 2] = mz;

  o_colors[3 * gi + 0] = 1.f / (1.f + expf(-colors_raw[3 * n + 0]));
  o_colors[3 * gi + 1] = 1.f / (1.f + expf(-colors_raw[3 * n + 1]));
  o_colors[3 * gi + 2] = 1.f / (1.f + expf(-colors_raw[3 * n + 2]));

  o_opacity[gi] = 1.f / (1.f + expf(-opacity_raw[n]));

  o_scales[3 * gi + 0] = fminf(fmaxf(expf(log_scales[3 * n + 0]), MIN_SCALE), MAX_SCALE);
  o_scales[3 * gi + 1] = fminf(fmaxf(expf(log_scales[3 * n + 1]), MIN_SCALE), MAX_SCALE);
  o_scales[3 * gi + 2] = fminf(fmaxf(expf(log_scales[3 * n + 2]), MIN_SCALE), MAX_SCALE);

  *(float4*)(o_rot + 4 * gi) = make_float4(rw, rxq, ryq, rzq);  // 16B-aligned b128 store
}

// ---------------------------------------------------------------------------
// kernel_launch
// ws layout (floats): Mpad B*J*16 (10240) | Vw B*V*3 (2.4M) | nrm B*V*3 (2.4M)
// total ~19.3 MB.
// ---------------------------------------------------------------------------
extern "C" void kernel_launch(void* const* d_in, const int* in_sizes, int n_in,
                              void* d_out, int out_size, void* d_ws, size_t ws_size,
                              hipStream_t stream) {
  const float* pose = (const float*)d_in[0];
  const float* trans = (const float*)d_in[1];
  const float* scale = (const float*)d_in[2];
  const float* v_template = (const float*)d_in[3];
  const float* skin_weights = (const float*)d_in[4];
  const float* joints_rest = (const float*)d_in[5];
  const float* position_offsets = (const float*)d_in[6];
  const float* colors_raw = (const float*)d_in[7];
  const float* opacity_raw = (const float*)d_in[8];
  const float* log_scales = (const float*)d_in[9];
  const float* quaternions = (const float*)d_in[10];
  const int* vertex_indices = (const int*)d_in[11];
  const int* faces = (const int*)d_in[12];

  float* ws = (float*)d_ws;
  float* Mpad = ws;                                  // B*J*16
  float* Vw = Mpad + (size_t)B_ * J_ * 16;           // B*V*3
  float* nrm = Vw + (size_t)B_ * V_ * 3;             // B*V*3

  float* out = (float*)d_out;
  float* o_means = out;                               // B*N*3
  float* o_colors = o_means + (size_t)B_ * N_ * 3;    // B*N*3
  float* o_opacity = o_colors + (size_t)B_ * N_ * 3;  // B*N
  float* o_scales = o_opacity + (size_t)B_ * N_;      // B*N*3
  float* o_rot = o_scales + (size_t)B_ * N_ * 3;      // B*N*4

  // 0) joint matrices (one wave)
  k0_joint_mats<<<1, 32, 0, stream>>>(pose, trans, scale, joints_rest, Mpad);

  // 0b) zero the normal accumulator
  int nz = B_ * V_ * 3;
  k_zero<<<(nz + 255) / 256, 256, 0, stream>>>(nrm, nz);

  // 1) WMMA skinning: one wave per (16-vertex tile, batch)
  dim3 g1((V_ + 15) / 16, B_);
  k1_skin_wmma<<<g1, 32, 0, stream>>>(skin_weights, Mpad, v_template, Vw);

  // 2) face-normal scatter
  dim3 g2((F_ + 255) / 256, B_);
  k2_face_normals<<<g2, 256, 0, stream>>>(Vw, faces, nrm);

  // 3) per-Gaussian streaming pass
  dim3 g3((N_ + 255) / 256, B_);
  k3_gauss<<<g3, 256, 0, stream>>>(Vw, nrm, position_offsets, colors_raw,
                                   opacity_raw, log_scales, quaternions,
                                   vertex_indices, o_means, o_colors, o_opacity,
                                   o_scales, o_rot);
}